// Net_8555574853921
// MI455X (gfx1250) — compile-verified
//
#include <hip/hip_runtime.h>
#include <cstdint>
#include <cstddef>

// ---------------------------------------------------------------------------
// GNN (edge-conditioned GAT x2) for MI455X / gfx1250.
// All dense layers run on V_WMMA_F32_16X16X4_F32 (fp32-exact tensor ops).
// BatchNorm handled with stats-pass + recompute so no [E,64] hidden tensor is
// ever materialized (keeps traffic in the 192MB L2 / far below HBM roofline).
// ---------------------------------------------------------------------------

#define NNODES 100000
#define NEDGES 1600000
#define BN_EPS 1e-5f

typedef __attribute__((ext_vector_type(2))) float v2f;
typedef __attribute__((ext_vector_type(8))) float v8f;

// ordered-uint encoding of -inf (host+device): bits(-inf)=0xFF800000, neg -> ~u
static constexpr unsigned ENC_NEG_INF = 0x007FFFFFu;

// ---------------- WMMA helpers (fp32 16x16x4, wave32 lane layouts) ----------

__device__ __forceinline__ v8f wmma4(v2f a, v2f b, v8f c) {
  // (neg_a, A, neg_b, B, c_mod, C, reuse_a, reuse_b)
  return __builtin_amdgcn_wmma_f32_16x16x4_f32(false, a, false, b, (short)0, c,
                                               false, false);
}

// A tile is [16 x K] in LDS, row stride sa. Lane layout (ISA 7.12.2):
//  VGPR0: lanes0-15 -> K=kb, lanes16-31 -> K=kb+2 ; VGPR1: K=kb+1 / kb+3
__device__ __forceinline__ v2f load_a(const float* A, int sa, int kc, int lane) {
  int row = lane & 15;
  int kb = (kc << 2) + ((lane >> 4) << 1);
  v2f a;
  a.x = A[row * sa + kb];
  a.y = A[row * sa + kb + 1];
  return a;
}

// B tile is [K x ncols] in LDS, row stride sb; 16 columns starting at c0.
//  VGPR0: lanes0-15 -> K=kb (N=lane), lanes16-31 -> K=kb+2 ; VGPR1: +1 / +3
__device__ __forceinline__ v2f load_b(const float* B, int sb, int kc, int c0,
                                      int lane) {
  int col = c0 + (lane & 15);
  int kb = (kc << 2) + ((lane >> 4) << 1);
  v2f b;
  b.x = B[kb * sb + col];
  b.y = B[(kb + 1) * sb + col];
  return b;
}

#define V8F_ZERO {0.f,0.f,0.f,0.f,0.f,0.f,0.f,0.f}

// ---------------- generic BN-MLP (lin->bn->relu->lin->bn->relu->lin) --------
// Recompute strategy: pass1 = layer1 stats; pass2 = layer1 recompute + layer2
// stats; pass3 = full recompute + output (optional residual / final relu).

template <int DH>
__global__ __launch_bounds__(256) void mlp_pass1(
    const float* __restrict__ in, int inStride, int Din, int DinP,
    const float* __restrict__ W1, const float* __restrict__ b1,
    float* __restrict__ stats, int nRows) {
  static_assert(DH == 32 || DH == 64, "");
  constexpr int COLT = DH / 16, ROWT = 8 / COLT, R = ROWT * 16;
  __shared__ float s_in[R * 20];
  __shared__ float s_w1[16 * DH];
  __shared__ float s_st[2 * DH];
  const int tid = threadIdx.x, lane = tid & 31, wid = tid >> 5;
  const int rt = wid / COLT, ct = wid % COLT;
  const int row0 = blockIdx.x * R;
  for (int i = tid; i < 2 * DH; i += 256) s_st[i] = 0.f;
  for (int i = tid; i < R * 16; i += 256) {
    int r = i >> 4, cc = i & 15, g = row0 + r;
    s_in[r * 20 + cc] = (g < nRows && cc < Din) ? in[(size_t)g * inStride + cc] : 0.f;
  }
  for (int i = tid; i < 16 * DH; i += 256) {
    int k = i / DH, j = i - k * DH;
    s_w1[i] = (k < Din) ? W1[(size_t)k * DH + j] : 0.f;
  }
  __syncthreads();
  v8f c = V8F_ZERO;
  {
    const float* A = &s_in[rt * 16 * 20];
    const int kch = DinP >> 2;
    for (int kc = 0; kc < kch; ++kc)
      c = wmma4(load_a(A, 20, kc, lane), load_b(s_w1, DH, kc, ct * 16, lane), c);
  }
  const int col = ct * 16 + (lane & 15);
  const float bb = b1[col];
  const int rbase = row0 + rt * 16 + ((lane >> 4) << 3);
  float ls = 0.f, lq = 0.f;
#pragma unroll
  for (int r = 0; r < 8; ++r) {
    float v = c[r] + bb;
    if (rbase + r < nRows) { ls += v; lq += v * v; }
  }
  atomicAdd(&s_st[col], ls);
  atomicAdd(&s_st[DH + col], lq);
  __syncthreads();
  for (int i = tid; i < 2 * DH; i += 256) atomicAdd(&stats[i], s_st[i]);
}

template <int DH>
__global__ __launch_bounds__(256) void mlp_pass2(
    const float* __restrict__ in, int inStride, int Din, int DinP,
    const float* __restrict__ W1, const float* __restrict__ b1,
    const float* __restrict__ ss1, const float* __restrict__ W2,
    const float* __restrict__ b2, float* __restrict__ stats, int nRows) {
  constexpr int COLT = DH / 16, ROWT = 8 / COLT, R = ROWT * 16, SH = DH + 4;
  __shared__ float s_in[R * 20];
  __shared__ float s_w1[16 * DH];
  __shared__ float s_w2[DH * DH];
  __shared__ float s_h1[R * SH];
  __shared__ float s_st[2 * DH];
  const int tid = threadIdx.x, lane = tid & 31, wid = tid >> 5;
  const int rt = wid / COLT, ct = wid % COLT;
  const int row0 = blockIdx.x * R;
  for (int i = tid; i < 2 * DH; i += 256) s_st[i] = 0.f;
  for (int i = tid; i < R * 16; i += 256) {
    int r = i >> 4, cc = i & 15, g = row0 + r;
    s_in[r * 20 + cc] = (g < nRows && cc < Din) ? in[(size_t)g * inStride + cc] : 0.f;
  }
  for (int i = tid; i < 16 * DH; i += 256) {
    int k = i / DH, j = i - k * DH;
    s_w1[i] = (k < Din) ? W1[(size_t)k * DH + j] : 0.f;
  }
  for (int i = tid; i < DH * DH; i += 256) s_w2[i] = W2[i];
  __syncthreads();
  const int col = ct * 16 + (lane & 15);
  v8f c = V8F_ZERO;
  {
    const float* A = &s_in[rt * 16 * 20];
    const int kch = DinP >> 2;
    for (int kc = 0; kc < kch; ++kc)
      c = wmma4(load_a(A, 20, kc, lane), load_b(s_w1, DH, kc, ct * 16, lane), c);
  }
  {
    const float bb = b1[col];
    const float sc = ss1[col], sh = ss1[DH + col];
    const int lrow0 = rt * 16 + ((lane >> 4) << 3);
#pragma unroll
    for (int r = 0; r < 8; ++r) {
      float v = (c[r] + bb) * sc + sh;
      v = v > 0.f ? v : 0.f;
      s_h1[(lrow0 + r) * SH + col] = v;
    }
  }
  __syncthreads();
  v8f c2 = V8F_ZERO;
  {
    const float* A2 = &s_h1[rt * 16 * SH];
    for (int kc = 0; kc < DH / 4; ++kc)
      c2 = wmma4(load_a(A2, SH, kc, lane), load_b(s_w2, DH, kc, ct * 16, lane), c2);
  }
  const float b2v = b2[col];
  const int rbase = row0 + rt * 16 + ((lane >> 4) << 3);
  float ls = 0.f, lq = 0.f;
#pragma unroll
  for (int r = 0; r < 8; ++r) {
    float v = c2[r] + b2v;
    if (rbase + r < nRows) { ls += v; lq += v * v; }
  }
  atomicAdd(&s_st[col], ls);
  atomicAdd(&s_st[DH + col], lq);
  __syncthreads();
  for (int i = tid; i < 2 * DH; i += 256) atomicAdd(&stats[i], s_st[i]);
}

template <int DH>
__global__ __launch_bounds__(256) void mlp_pass3(
    const float* __restrict__ in, int inStride, int Din, int DinP,
    const float* __restrict__ W1, const float* __restrict__ b1,
    const float* __restrict__ ss1, const float* __restrict__ W2,
    const float* __restrict__ b2, const float* __restrict__ ss2,
    const float* __restrict__ W3, const float* __restrict__ b3,
    float* __restrict__ out, int outStride, int Dout,
    const float* __restrict__ resid, int residStride, int doRelu, int nRows) {
  constexpr int COLT = DH / 16, ROWT = 8 / COLT, R = ROWT * 16, SH = DH + 4;
  __shared__ float s_in[R * 20];
  __shared__ float s_w1[16 * DH];
  __shared__ float s_w2[DH * DH];
  __shared__ float s_w3[DH * 32];
  __shared__ float s_h1[R * SH];
  __shared__ float s_h2[R * SH];
  const int tid = threadIdx.x, lane = tid & 31, wid = tid >> 5;
  const int rt = wid / COLT, ct = wid % COLT;
  const int row0 = blockIdx.x * R;
  for (int i = tid; i < R * 16; i += 256) {
    int r = i >> 4, cc = i & 15, g = row0 + r;
    s_in[r * 20 + cc] = (g < nRows && cc < Din) ? in[(size_t)g * inStride + cc] : 0.f;
  }
  for (int i = tid; i < 16 * DH; i += 256) {
    int k = i / DH, j = i - k * DH;
    s_w1[i] = (k < Din) ? W1[(size_t)k * DH + j] : 0.f;
  }
  for (int i = tid; i < DH * DH; i += 256) s_w2[i] = W2[i];
  for (int i = tid; i < DH * 32; i += 256) {
    int k = i >> 5, j = i & 31;
    s_w3[i] = (j < Dout) ? W3[(size_t)k * Dout + j] : 0.f;
  }
  __syncthreads();
  const int col = ct * 16 + (lane & 15);
  const int lrow0 = rt * 16 + ((lane >> 4) << 3);
  v8f c = V8F_ZERO;
  {
    const float* A = &s_in[rt * 16 * 20];
    const int kch = DinP >> 2;
    for (int kc = 0; kc < kch; ++kc)
      c = wmma4(load_a(A, 20, kc, lane), load_b(s_w1, DH, kc, ct * 16, lane), c);
  }
  {
    const float bb = b1[col];
    const float sc = ss1[col], sh = ss1[DH + col];
#pragma unroll
    for (int r = 0; r < 8; ++r) {
      float v = (c[r] + bb) * sc + sh;
      s_h1[(lrow0 + r) * SH + col] = v > 0.f ? v : 0.f;
    }
  }
  __syncthreads();
  v8f c2 = V8F_ZERO;
  {
    const float* A2 = &s_h1[rt * 16 * SH];
    for (int kc = 0; kc < DH / 4; ++kc)
      c2 = wmma4(load_a(A2, SH, kc, lane), load_b(s_w2, DH, kc, ct * 16, lane), c2);
  }
  {
    const float bb = b2[col];
    const float sc = ss2[col], sh = ss2[DH + col];
#pragma unroll
    for (int r = 0; r < 8; ++r) {
      float v = (c2[r] + bb) * sc + sh;
      s_h2[(lrow0 + r) * SH + col] = v > 0.f ? v : 0.f;
    }
  }
  __syncthreads();
  const int DoutT = (Dout + 15) >> 4;
  if (ct < DoutT) {
    v8f c3 = V8F_ZERO;
    const float* A3 = &s_h2[rt * 16 * SH];
    for (int kc = 0; kc < DH / 4; ++kc)
      c3 = wmma4(load_a(A3, SH, kc, lane), load_b(s_w3, 32, kc, ct * 16, lane), c3);
    const int col3 = ct * 16 + (lane & 15);
    if (col3 < Dout) {
      const float b3v = b3[col3];
      const int rbase = row0 + lrow0;
#pragma unroll
      for (int r = 0; r < 8; ++r) {
        int g = rbase + r;
        if (g < nRows) {
          float v = c3[r] + b3v;
          if (resid) v += resid[(size_t)g * residStride + col3];
          if (doRelu) v = v > 0.f ? v : 0.f;
          out[(size_t)g * outStride + col3] = v;
        }
      }
    }
  }
}

// ---------------- small utility kernels -------------------------------------

__global__ void fill_f32(float* p, float v, long n) {
  long i = (long)blockIdx.x * blockDim.x + threadIdx.x;
  if (i < n) p[i] = v;
}
__global__ void fill_u32(unsigned* p, unsigned v, long n) {
  long i = (long)blockIdx.x * blockDim.x + threadIdx.x;
  if (i < n) p[i] = v;
}

// stats -> (scale, shift): y = x*scale + shift with scale=g*rsqrt(var+eps)
__global__ void bn_finalize(const float* __restrict__ stats,
                            const float* __restrict__ g,
                            const float* __restrict__ be, float* __restrict__ ss,
                            int D, float invN) {
  int i = threadIdx.x;
  if (i < D) {
    float mu = stats[i] * invN;
    float var = stats[D + i] * invN - mu * mu;
    float r = rsqrtf(var + BN_EPS);
    float sc = g[i] * r;
    ss[i] = sc;
    ss[D + i] = be[i] - mu * sc;
  }
}

__global__ void bn_stats(const float* __restrict__ x, int D, int preRelu,
                         float* __restrict__ stats, int n) {
  __shared__ float s[64];  // 2*Dmax (D<=32)
  const int tid = threadIdx.x;
  for (int i = tid; i < 2 * D; i += blockDim.x) s[i] = 0.f;
  __syncthreads();
  int node = blockIdx.x * blockDim.x + tid;
  if (node < n) {
    for (int j = 0; j < D; ++j) {
      float v = x[(size_t)node * D + j];
      if (preRelu) v = v > 0.f ? v : 0.f;
      atomicAdd(&s[j], v);
      atomicAdd(&s[D + j], v * v);
    }
  }
  __syncthreads();
  for (int i = tid; i < 2 * D; i += blockDim.x) atomicAdd(&stats[i], s[i]);
}

__global__ void bn_apply(const float* __restrict__ x, int D, int preRelu,
                         const float* __restrict__ ss, float* __restrict__ y,
                         long n) {
  long i = (long)blockIdx.x * blockDim.x + threadIdx.x;
  if (i >= n * D) return;
  int c = (int)(i % D);
  float v = x[i];
  if (preRelu) v = v > 0.f ? v : 0.f;
  y[i] = v * ss[c] + ss[D + c];
}

__global__ void add_bias(float* __restrict__ x, const float* __restrict__ b,
                         int D, long n) {
  long i = (long)blockIdx.x * blockDim.x + threadIdx.x;
  if (i < n * D) x[i] += b[i % D];
}

// ---------------- GAT kernels ----------------------------------------------

__device__ __forceinline__ unsigned fenc(float f) {
  unsigned u = __float_as_uint(f);
  return (u & 0x80000000u) ? ~u : (u | 0x80000000u);
}
__device__ __forceinline__ float fdec(unsigned u) {
  unsigned v = (u & 0x80000000u) ? (u & 0x7FFFFFFFu) : ~u;
  return __uint_as_float(v);
}

// h0 = x @ W ; sdot = h0.as ; ddot = h0.ad   (din<=7, dout<=16)
__global__ void gat_node(const float* __restrict__ x, int inStride, int din,
                         int dout, const float* __restrict__ W,
                         const float* __restrict__ as_, const float* __restrict__ ad_,
                         float* __restrict__ h0, float* __restrict__ sdot,
                         float* __restrict__ ddot, int n) {
  int node = blockIdx.x * blockDim.x + threadIdx.x;
  if (node >= n) return;
  float xv[7];
  for (int i = 0; i < din; ++i) xv[i] = x[(size_t)node * inStride + i];
  float s = 0.f, d = 0.f;
  for (int j = 0; j < dout; ++j) {
    float hv = 0.f;
    for (int i = 0; i < din; ++i) hv += xv[i] * W[i * dout + j];
    h0[(size_t)node * dout + j] = hv;
    s += hv * as_[j];
    d += hv * ad_[j];
  }
  sdot[node] = s;
  ddot[node] = d;
}

// wae = We @ ae  (edge_attr @ We) . ae == edge_attr . (We @ ae)
__global__ void compute_wae(const float* __restrict__ We,
                            const float* __restrict__ ae, float* __restrict__ wae,
                            int dedge, int dout) {
  int i = threadIdx.x;
  if (i < dedge) {
    float s = 0.f;
    for (int j = 0; j < dout; ++j) s += We[i * dout + j] * ae[j];
    wae[i] = s;
  }
}

__global__ void gat_alpha(const float* __restrict__ ereg, int dedge,
                          const float* __restrict__ wae,
                          const float* __restrict__ sdot,
                          const float* __restrict__ ddot,
                          const int* __restrict__ src, const int* __restrict__ dst,
                          float* __restrict__ alpha, unsigned* __restrict__ menc,
                          long E) {
  long e = (long)blockIdx.x * blockDim.x + threadIdx.x;
  if (e >= E) return;
  int s = src[e], d = dst[e];
  float a = sdot[s] + ddot[d];
  for (int i = 0; i < dedge; ++i) a += ereg[e * dedge + i] * wae[i];
  a = a >= 0.f ? a : 0.2f * a;  // leaky_relu(0.2)
  alpha[e] = a;
  atomicMax(&menc[d], fenc(a));
}

__global__ void decode_max(const unsigned* __restrict__ menc,
                           float* __restrict__ mf, int n) {
  int i = blockIdx.x * blockDim.x + threadIdx.x;
  if (i >= n) return;
  float v = fdec(menc[i]);
  // where(isfinite(m), m, 0)
  mf[i] = ((__float_as_uint(v) & 0x7F800000u) != 0x7F800000u) ? v : 0.f;
}

__global__ void gat_exp(float* __restrict__ alpha, const int* __restrict__ dst,
                        const float* __restrict__ mf, float* __restrict__ ssum,
                        long E) {
  long e = (long)blockIdx.x * blockDim.x + threadIdx.x;
  if (e >= E) return;
  int d = dst[e];
  float a = expf(alpha[e] - mf[d]);
  alpha[e] = a;
  atomicAdd(&ssum[d], a);
}

__global__ void gat_scatter(const float* __restrict__ a, const int* __restrict__ src,
                            const int* __restrict__ dst,
                            const float* __restrict__ ssum,
                            const float* __restrict__ h0, int dout,
                            float* __restrict__ aggr, long E) {
  long e = (long)blockIdx.x * blockDim.x + threadIdx.x;
  if (e >= E) return;
  int s = src[e], d = dst[e];
  float w = a[e] / (ssum[d] + 1e-16f);
  for (int j = 0; j < dout; ++j)
    atomicAdd(&aggr[(size_t)d * dout + j], w * h0[(size_t)s * dout + j]);
}

// ---------------- edge input builders --------------------------------------

__global__ void build_reg1(const float* __restrict__ pts,
                           const float* __restrict__ x,
                           const int* __restrict__ src, const int* __restrict__ dst,
                           float* __restrict__ out4, long E) {
  long e = (long)blockIdx.x * blockDim.x + threadIdx.x;
  if (e >= E) return;
  int s = src[e], d = dst[e];
  out4[e * 4 + 0] = pts[s * 3 + 0] - pts[d * 3 + 0];
  out4[e * 4 + 1] = pts[s * 3 + 1] - pts[d * 3 + 1];
  out4[e * 4 + 2] = pts[s * 3 + 2] - pts[d * 3 + 2];
  out4[e * 4 + 3] = x[(size_t)d * 7 + 3];
}

__global__ void build_reg2(const float* __restrict__ pts,
                           const float* __restrict__ h7,
                           const int* __restrict__ src, const int* __restrict__ dst,
                           float* __restrict__ out10, long E) {
  long e = (long)blockIdx.x * blockDim.x + threadIdx.x;
  if (e >= E) return;
  int s = src[e], d = dst[e];
  out10[e * 10 + 0] = pts[s * 3 + 0] - pts[d * 3 + 0];
  out10[e * 10 + 1] = pts[s * 3 + 1] - pts[d * 3 + 1];
  out10[e * 10 + 2] = pts[s * 3 + 2] - pts[d * 3 + 2];
  for (int j = 0; j < 7; ++j) out10[e * 10 + 3 + j] = h7[(size_t)d * 7 + j];
}

__global__ void final_sigmoid(const float* __restrict__ h, const float* __restrict__ W,
                              const float* __restrict__ b, float* __restrict__ out,
                              int n) {
  int node = blockIdx.x * blockDim.x + threadIdx.x;
  if (node >= n) return;
  float z = b[0];
  for (int c = 0; c < 32; ++c) z += h[(size_t)node * 32 + c] * W[c];
  out[node] = 1.f / (1.f + expf(-z));
}

// ---------------- host orchestration ---------------------------------------

static inline int cdiv(long a, int b) { return (int)((a + b - 1) / b); }

template <int DH>
static void run_mlp(hipStream_t st, const float* in, int inStride, int Din,
                    int DinP, const float* const* p10, float* statsA, float* ssA,
                    float* statsB, float* ssB, float* out, int outStride,
                    int Dout, const float* resid, int residStride, int doRelu,
                    int nRows) {
  constexpr int R = (DH == 64) ? 32 : 64;
  const int nb = cdiv(nRows, R);
  const float invN = 1.0f / (float)nRows;
  fill_f32<<<1, 256, 0, st>>>(statsA, 0.f, 2 * DH);
  mlp_pass1<DH><<<nb, 256, 0, st>>>(in, inStride, Din, DinP, p10[0], p10[1],
                                    statsA, nRows);
  bn_finalize<<<1, DH, 0, st>>>(statsA, p10[2], p10[3], ssA, DH, invN);
  fill_f32<<<1, 256, 0, st>>>(statsB, 0.f, 2 * DH);
  mlp_pass2<DH><<<nb, 256, 0, st>>>(in, inStride, Din, DinP, p10[0], p10[1], ssA,
                                    p10[4], p10[5], statsB, nRows);
  bn_finalize<<<1, DH, 0, st>>>(statsB, p10[6], p10[7], ssB, DH, invN);
  mlp_pass3<DH><<<nb, 256, 0, st>>>(in, inStride, Din, DinP, p10[0], p10[1], ssA,
                                    p10[4], p10[5], ssB, p10[8], p10[9], out,
                                    outStride, Dout, resid, residStride, doRelu,
                                    nRows);
}

extern "C" void kernel_launch(void* const* d_in, const int* in_sizes, int n_in,
                              void* d_out, int out_size, void* d_ws,
                              size_t ws_size, hipStream_t stream) {
  const int N = NNODES;
  const long E = NEDGES;
  const int NB = cdiv(N, 256);
  const int EB = cdiv(E, 256);

  auto F = [&](int i) -> const float* { return (const float*)d_in[i]; };

  // Input order assumption: setup_inputs() insertion order, params flattened
  // depth-first in _make_params insertion order (leaf dicts W,b,g,be).
  const float* x = F(0);                     // [N,7]
  const int* ei = (const int*)d_in[1];       // [2,E] (int32)
  const int* src = ei;
  const int* dst = ei + E;
  // MLP leaf bundles: base -> {W1,b1,g1,be1,W2,b2,g2,be2,W3,b3}
  auto mlpP = [&](int base, const float** p) {
    p[0] = F(base + 0); p[1] = F(base + 1); p[2] = F(base + 2); p[3] = F(base + 3);
    p[4] = F(base + 4); p[5] = F(base + 5); p[6] = F(base + 6); p[7] = F(base + 7);
    p[8] = F(base + 8); p[9] = F(base + 9);
  };
  const float *m11[10], *m14[10], *m12[10], *m17[10], *m18[10], *m13[10];
  mlpP(2, m11);    // mlp1_1 [3,64,64,3]
  mlpP(14, m14);   // mlp1_4 [4,64,64,7]
  const float* c1_W = F(26); const float* c1_We = F(27); const float* c1_as = F(28);
  const float* c1_ad = F(29); const float* c1_ae = F(30); const float* c1_b = F(31);
  const float* bn11_g = F(32); const float* bn11_be = F(33);
  mlpP(34, m12);   // mlp1_2 [7,64,64,7]
  const float* bn12_g = F(46); const float* bn12_be = F(47);
  mlpP(48, m17);   // mlp1_7 [3,32,32,3]
  mlpP(60, m18);   // mlp1_8 [10,32,32,7]
  const float* c2_W = F(72); const float* c2_We = F(73); const float* c2_as = F(74);
  const float* c2_ad = F(75); const float* c2_ae = F(76); const float* c2_b = F(77);
  const float* bn13_g = F(78); const float* bn13_be = F(79);
  mlpP(80, m13);   // mlp1_3 [16,64,64,32]
  const float* bn14_g = F(92); const float* bn14_be = F(93);
  const float* lin_W = F(94); const float* lin_b = F(95);

  // --- workspace carve-up (≈230MB) ---
  char* base = (char*)d_ws;
  size_t off = 0;
  auto allocB = [&](size_t bytes) -> void* {
    void* p = base + off;
    off += (bytes + 255) & ~(size_t)255;
    return p;
  };
  float* pts1 = (float*)allocB((size_t)N * 3 * 4);
  float* pts2 = (float*)allocB((size_t)N * 3 * 4);
  float* reg1in = (float*)allocB((size_t)E * 4 * 4);
  float* reg1out = (float*)allocB((size_t)E * 7 * 4);
  float* reg2in = (float*)allocB((size_t)E * 10 * 4);
  float* reg2out = (float*)allocB((size_t)E * 7 * 4);
  float* alphaB = (float*)allocB((size_t)E * 4);
  float* h0 = (float*)allocB((size_t)N * 16 * 4);
  float* sdot = (float*)allocB((size_t)N * 4);
  float* ddot = (float*)allocB((size_t)N * 4);
  unsigned* menc = (unsigned*)allocB((size_t)N * 4);
  float* mf = (float*)allocB((size_t)N * 4);
  float* ssum = (float*)allocB((size_t)N * 4);
  float* aggr = (float*)allocB((size_t)N * 16 * 4);
  float* nodeA = (float*)allocB((size_t)N * 32 * 4);
  float* nodeB = (float*)allocB((size_t)N * 32 * 4);
  float* statsA = (float*)allocB(128 * 4);
  float* ssA = (float*)allocB(128 * 4);
  float* statsB = (float*)allocB(128 * 4);
  float* ssB = (float*)allocB(128 * 4);
  float* wae = (float*)allocB(16 * 4);
  (void)ws_size; (void)n_in; (void)in_sizes; (void)out_size;

  auto run_gat = [&](const float* nodeIn, int inStride, int dout,
                     const float* W, const float* We, const float* as_,
                     const float* ad_, const float* ae_, const float* bias,
                     const float* ereg, int dedge, float* outAggr) {
    gat_node<<<NB, 256, 0, stream>>>(nodeIn, inStride, 7, dout, W, as_, ad_, h0,
                                     sdot, ddot, N);
    compute_wae<<<1, 32, 0, stream>>>(We, ae_, wae, dedge, dout);
    fill_u32<<<NB, 256, 0, stream>>>(menc, ENC_NEG_INF, N);
    fill_f32<<<NB, 256, 0, stream>>>(ssum, 0.f, N);
    fill_f32<<<cdiv((long)N * dout, 256), 256, 0, stream>>>(outAggr, 0.f,
                                                            (long)N * dout);
    gat_alpha<<<EB, 256, 0, stream>>>(ereg, dedge, wae, sdot, ddot, src, dst,
                                      alphaB, menc, E);
    decode_max<<<NB, 256, 0, stream>>>(menc, mf, N);
    gat_exp<<<EB, 256, 0, stream>>>(alphaB, dst, mf, ssum, E);
    gat_scatter<<<EB, 256, 0, stream>>>(alphaB, src, dst, ssum, h0, dout,
                                        outAggr, E);
    add_bias<<<cdiv((long)N * dout, 256), 256, 0, stream>>>(outAggr, bias, dout,
                                                            (long)N);
  };

  auto run_bn = [&](const float* xin, int D, int preRelu, const float* g,
                    const float* be, float* yout) {
    fill_f32<<<1, 256, 0, stream>>>(statsA, 0.f, 2 * D);
    bn_stats<<<NB, 256, 0, stream>>>(xin, D, preRelu, statsA, N);
    bn_finalize<<<1, D, 0, stream>>>(statsA, g, be, ssA, D, 1.0f / (float)N);
    bn_apply<<<cdiv((long)N * D, 256), 256, 0, stream>>>(xin, D, preRelu, ssA,
                                                         yout, (long)N);
  };

  // ---- stage 1: registration nets (node MLPs + residual) ----
  // pts1 = x[:,0:3] + mlp1_1(x[:,0:3])
  run_mlp<64>(stream, x, 7, 3, 4, m11, statsA, ssA, statsB, ssB, pts1, 3, 3, x,
              7, 0, N);
  // pts2 = x[:,0:3] + mlp1_7(x[:,0:3])
  run_mlp<32>(stream, x, 7, 3, 4, m17, statsA, ssA, statsB, ssB, pts2, 3, 3, x,
              7, 0, N);

  // ---- stage 2: edge features 1 + mlp1_4 (heavy, WMMA) ----
  build_reg1<<<EB, 256, 0, stream>>>(pts1, x, src, dst, reg1in, E);
  run_mlp<64>(stream, reg1in, 4, 4, 4, m14, statsA, ssA, statsB, ssB, reg1out, 7,
              7, nullptr, 0, /*relu=*/1, (int)E);

  // ---- stage 3: GAT conv1_1 + bn1_1 ----
  run_gat(x, 7, 7, c1_W, c1_We, c1_as, c1_ad, c1_ae, c1_b, reg1out, 7, aggr);
  run_bn(aggr, 7, 0, bn11_g, bn11_be, nodeA);  // nodeA = h [N,7]

  // ---- stage 4: mlp1_2 then relu+bn1_2 ----
  run_mlp<64>(stream, nodeA, 7, 7, 8, m12, statsA, ssA, statsB, ssB, nodeB, 7, 7,
              nullptr, 0, 0, N);
  run_bn(nodeB, 7, /*preRelu=*/1, bn12_g, bn12_be, nodeA);  // nodeA = h [N,7]

  // ---- stage 5: edge features 2 + mlp1_8 ----
  build_reg2<<<EB, 256, 0, stream>>>(pts2, nodeA, src, dst, reg2in, E);
  run_mlp<32>(stream, reg2in, 10, 10, 12, m18, statsA, ssA, statsB, ssB, reg2out,
              7, 7, nullptr, 0, /*relu=*/1, (int)E);

  // ---- stage 6: GAT conv1_2 + bn1_3 ----
  run_gat(nodeA, 7, 16, c2_W, c2_We, c2_as, c2_ad, c2_ae, c2_b, reg2out, 7, aggr);
  run_bn(aggr, 16, 0, bn13_g, bn13_be, nodeB);  // nodeB = h [N,16]

  // ---- stage 7: mlp1_3 then relu+bn1_4, sigmoid head ----
  run_mlp<64>(stream, nodeB, 16, 16, 16, m13, statsA, ssA, statsB, ssB, nodeA,
              32, 32, nullptr, 0, 0, N);
  run_bn(nodeA, 32, /*preRelu=*/1, bn14_g, bn14_be, nodeB);  // nodeB [N,32]
  final_sigmoid<<<NB, 256, 0, stream>>>(nodeB, lin_W, lin_b, (float*)d_out, N);
}